// GCNII_23089744183809
// MI455X (gfx1250) — compile-verified
//
#include <hip/hip_runtime.h>
#include <math.h>

// GCNII forward for MI455X (gfx1250, wave32).
// Bandwidth/atomic-bound workload; GEMMs done with V_WMMA_F32_16X16X4_F32
// (exact f32 math, compute is trivially small vs. the matrix-core ceiling).

typedef __attribute__((ext_vector_type(2))) float v2f;
typedef __attribute__((ext_vector_type(8))) float v8f;

#define DF 128   // feature dim (DIN == DOUT == 128)

// out[row,col] = relu( coefW * (A @ W)[row,col] + coefC * A[row,col] + bias[col] )
// Block = 256 threads = 8 waves. Each block owns a 16-row tile; wave w owns
// columns [16w, 16w+16). K-loop: 32 steps of 16x16x4 f32 WMMA.
__global__ __launch_bounds__(256) void gcnii_gemm_kernel(
    const float* __restrict__ A,      // [n, 128]
    const float* __restrict__ W,      // [128, 128] row-major (k, n)
    const float* __restrict__ bias,   // [128] or nullptr
    float* __restrict__ out,          // [n, 128]
    float* __restrict__ out2,         // optional second copy or nullptr
    int n, float coefC, float coefW)
{
    const int wave    = threadIdx.x >> 5;   // 0..7
    const int lane    = threadIdx.x & 31;
    const int rowBase = blockIdx.x * 16;
    if (rowBase >= n) return;               // block-uniform: EXEC stays all-1s for WMMA
    const int colBase = wave * 16;
    const int halfSel = lane >> 4;          // 0: lanes 0-15, 1: lanes 16-31
    const int lane16  = lane & 15;

    // A-frag row for this lane (clamped for safety on ragged tails)
    int arowIdx = rowBase + lane16;
    if (arowIdx >= n) arowIdx = n - 1;
    const float* __restrict__ arow = A + (size_t)arowIdx * DF;

    v8f c = {0.f, 0.f, 0.f, 0.f, 0.f, 0.f, 0.f, 0.f};

    // ISA 16x4 f32 A layout: lanes 0-15 hold K={k0,k0+1}, lanes 16-31 hold K={k0+2,k0+3}
    // B (4x16) mirrored: same K split, N striped across lane16 within a VGPR.
#pragma unroll 4
    for (int k0 = 0; k0 < DF; k0 += 4) {
        const int ka = k0 + halfSel * 2;
        v2f a, b;
        a.x = arow[ka];
        a.y = arow[ka + 1];
        b.x = W[(size_t)ka       * DF + colBase + lane16];
        b.y = W[(size_t)(ka + 1) * DF + colBase + lane16];
        c = __builtin_amdgcn_wmma_f32_16x16x4_f32(
                /*neg_a=*/false, a, /*neg_b=*/false, b,
                /*c_mod=*/(short)0, c, /*reuse_a=*/false, /*reuse_b=*/false);
    }

    const int   col  = colBase + lane16;
    const float bval = bias ? bias[col] : 0.0f;

    // C/D layout: VGPR j -> M = j (lanes 0-15) / j+8 (lanes 16-31), N = lane16
#pragma unroll
    for (int j = 0; j < 8; ++j) {
        const int row = rowBase + j + halfSel * 8;
        if (row >= n) continue;
        float v = coefW * c[j] + bval;
        if (coefC != 0.0f) v += coefC * A[(size_t)row * DF + col];  // residual term (L2 hit)
        v = v > 0.0f ? v : 0.0f;                                    // relu
        out[(size_t)row * DF + col] = v;
        if (out2) out2[(size_t)row * DF + col] = v;
    }
}

// SpMM scatter: agg[dst] += h[src] * w. One wave per edge (grid-stride over
// edges); each lane covers 4 consecutive features via a float4 gather and
// 4 global f32 atomic adds. h fits in the 192MB L2 so gathers are cheap.
__global__ __launch_bounds__(256) void gcnii_spmm_kernel(
    const float* __restrict__ h,          // [n, 128]
    const long long* __restrict__ srcIdx, // [E]
    const long long* __restrict__ dstIdx, // [E]
    const float* __restrict__ ew,         // [E]
    float* __restrict__ agg,              // [n, 128], pre-zeroed
    int nEdges)
{
    const int lane   = threadIdx.x & 31;
    const int waveId = blockIdx.x * (blockDim.x >> 5) + (threadIdx.x >> 5);
    const int nWaves = gridDim.x * (blockDim.x >> 5);
    const int f      = lane * 4;          // 32 lanes x 4 floats = 128 features

    for (int e = waveId; e < nEdges; e += nWaves) {
        const long long s = srcIdx[e];
        const long long d = dstIdx[e];
        const float     w = ew[e];
        const float4 hv = *(const float4*)(h + (size_t)s * DF + f);
        float* ap = agg + (size_t)d * DF + f;
        atomicAdd(ap + 0, hv.x * w);
        atomicAdd(ap + 1, hv.y * w);
        atomicAdd(ap + 2, hv.z * w);
        atomicAdd(ap + 3, hv.w * w);
    }
}

// m = (1-ALPHA)*agg + ALPHA*x0, vectorized float4 grid-stride.
__global__ __launch_bounds__(256) void gcnii_mix_kernel(
    const float* __restrict__ agg,
    const float* __restrict__ x0,
    float* __restrict__ m,
    int total4)
{
    int i      = blockIdx.x * blockDim.x + threadIdx.x;
    int stride = gridDim.x * blockDim.x;
    for (; i < total4; i += stride) {
        float4 a = ((const float4*)agg)[i];
        float4 x = ((const float4*)x0)[i];
        float4 r;
        r.x = 0.9f * a.x + 0.1f * x.x;
        r.y = 0.9f * a.y + 0.1f * x.y;
        r.z = 0.9f * a.z + 0.1f * x.z;
        r.w = 0.9f * a.w + 0.1f * x.w;
        ((float4*)m)[i] = r;
    }
}

extern "C" void kernel_launch(void* const* d_in, const int* in_sizes, int n_in,
                              void* d_out, int out_size, void* d_ws, size_t ws_size,
                              hipStream_t stream) {
    // setup_inputs() dict order: x, edge_weight, W_lin, b_lin, W_convs, edge_index
    const float*     x       = (const float*)d_in[0];
    const float*     ew      = (const float*)d_in[1];
    const float*     W_lin   = (const float*)d_in[2];
    const float*     b_lin   = (const float*)d_in[3];
    const float*     W_convs = (const float*)d_in[4];
    const long long* eidx    = (const long long*)d_in[5];   // int64 [2, E]

    const int N = in_sizes[0] / DF;
    const int E = in_sizes[1];
    const int L = in_sizes[4] / (DF * DF);
    float* out = (float*)d_out;                              // h lives here

    // Workspace layout: x0 | agg | m   (3 * N * 128 floats = ~76.8 MB)
    float* x0  = (float*)d_ws;
    float* agg = x0  + (size_t)N * DF;
    float* m   = agg + (size_t)N * DF;

    const long long* srcIdx = eidx;
    const long long* dstIdx = eidx + E;

    const int rowTiles   = (N + 15) / 16;     // 3125 blocks, 8 waves each
    const int spmmBlocks = 2048;              // 16384 waves, ~49 edges each
    const int mixBlocks  = 1024;
    const int total4     = N * DF / 4;

    // x0 = relu(x @ W_lin + b);  h = x0  (written into d_out too)
    gcnii_gemm_kernel<<<rowTiles, 256, 0, stream>>>(
        x, W_lin, b_lin, x0, out, N, /*coefC=*/0.0f, /*coefW=*/1.0f);

    for (int l = 0; l < L; ++l) {
        hipMemsetAsync(agg, 0, (size_t)N * DF * sizeof(float), stream);
        gcnii_spmm_kernel<<<spmmBlocks, 256, 0, stream>>>(
            out, srcIdx, dstIdx, ew, agg, E);
        gcnii_mix_kernel<<<mixBlocks, 256, 0, stream>>>(agg, x0, m, total4);

        const float beta = logf(0.5f / (float)(l + 1) + 1.0f);  // log(theta/l + 1)
        gcnii_gemm_kernel<<<rowTiles, 256, 0, stream>>>(
            m, W_convs + (size_t)l * DF * DF, /*bias=*/nullptr,
            out, /*out2=*/nullptr, N, /*coefC=*/1.0f - beta, /*coefW=*/beta);
    }
}